// myGAT_4157528342731
// MI455X (gfx1250) — compile-verified
//
#include <hip/hip_runtime.h>
#include <math.h>

#define IN_DIM   256
#define HID      32
#define HEADS    4
#define D1       128   // HEADS*HID
#define OUT_DIM  40
#define OUT_PAD  48    // OUT_DIM rounded up to 16
#define NEG_SLOPE 0.2f

typedef __attribute__((ext_vector_type(2))) float v2f;
typedef __attribute__((ext_vector_type(8))) float v8f;

// D = A(16x4) * B(4x16) + C, f32, wave32.
__device__ __forceinline__ v8f wmma4(v2f a, v2f b, v8f c) {
  return __builtin_amdgcn_wmma_f32_16x16x4_f32(false, a, false, b, (short)0, c,
                                               false, false);
}

__device__ __forceinline__ float lrelu(float x) { return x >= 0.f ? x : NEG_SLOPE * x; }

// monotonic float<->uint map so atomicMax(u32) implements float max
__device__ __forceinline__ unsigned f2o(float f) {
  unsigned u = __float_as_uint(f);
  return (u & 0x80000000u) ? ~u : (u | 0x80000000u);
}
__device__ __forceinline__ float o2f(unsigned v) {
  unsigned u = (v & 0x80000000u) ? (v ^ 0x80000000u) : ~v;
  return __uint_as_float(u);
}

// ---------------- init ----------------
__global__ void k_zero(float* p, long long n) {
  long long i = (long long)blockIdx.x * blockDim.x + threadIdx.x;
  if (i < n) p[i] = 0.f;
}

// ---------------- pad B [K x NC] -> [K x NCpad] (zeros in pad cols) ----------------
__global__ void k_padB(const float* __restrict__ B, float* __restrict__ Bp,
                       int K, int NC, int NCpad) {
  int i = blockIdx.x * blockDim.x + threadIdx.x;
  if (i >= K * NCpad) return;
  int k = i / NCpad, n = i - k * NCpad;
  Bp[i] = (n < NC) ? B[k * NC + n] : 0.f;
}

// ---------------- WMMA GEMM: C[M x NC] = A[M x K] * Bp[K x NCpad] ----------------
// K, NCpad compile-time so B's second-row load folds into an immediate offset.
// grid.x = M/16 row tiles; each wave of the block owns one 16-col tile.
template <int K, int NCPAD>
__global__ void k_gemm_wmma(const float* __restrict__ A, const float* __restrict__ Bp,
                            float* __restrict__ C, int NC, int Mrows) {
  int lane = threadIdx.x & 31;
  int wv   = threadIdx.x >> 5;
  int row0 = blockIdx.x * 16;
  int col0 = wv * 16;
  int m    = lane & 15;
  int koff = (lane < 16) ? 0 : 2;   // A/B fragment K offset per half-wave
  int n    = col0 + (lane & 15);

  int rowa = row0 + m; if (rowa >= Mrows) rowa = Mrows - 1;  // safe clamp (N%16==0 anyway)

  // 32-bit offsets, literal increments -> saddr + voffset + imm-offset addressing
  int aoff = rowa * K + koff;       // fits: 100000*256 < 2^31
  int boff = koff * NCPAD + n;

  v8f acc = {};
#pragma unroll 8
  for (int k = 0; k < K; k += 4) {
    v2f a = *(const v2f*)(A + aoff);           // contiguous 8B: K {0,1} / {2,3}
    v2f b;
    b.x = Bp[boff];                            // row kb,   col n
    b.y = Bp[boff + NCPAD];                    // row kb+1, col n (imm offset)
    acc = wmma4(a, b, acc);
    aoff += 4;
    boff += 4 * NCPAD;
  }

  int mst = (lane < 16) ? 0 : 8;
  int col = col0 + (lane & 15);
  for (int r = 0; r < 8; ++r) {
    int row = row0 + r + mst;
    if (col < NC && row < Mrows) C[(long long)row * NC + col] = acc[r];
  }
}

// ---------------- layer-1 attention coefficients: one wave per node ----------------
__global__ void k_alpha1(const float* __restrict__ h1, const float* __restrict__ a_src,
                         const float* __restrict__ a_dst, float* __restrict__ as1,
                         float* __restrict__ ad1, int N) {
  int wv = (int)(((long long)blockIdx.x * blockDim.x + threadIdx.x) >> 5);
  int lane = threadIdx.x & 31;
  if (wv >= N) return;
  float4 hv = ((const float4*)(h1 + (long long)wv * D1))[lane];
  int head  = lane >> 3;
  int cbase = (lane & 7) * 4;
  const float* asr = a_src + head * HID + cbase;
  const float* adr = a_dst + head * HID + cbase;
  float vs = hv.x * asr[0] + hv.y * asr[1] + hv.z * asr[2] + hv.w * asr[3];
  float vd = hv.x * adr[0] + hv.y * adr[1] + hv.z * adr[2] + hv.w * adr[3];
  for (int off = 1; off < 8; off <<= 1) { vs += __shfl_xor(vs, off); vd += __shfl_xor(vd, off); }
  if ((lane & 7) == 0) { as1[wv * 4 + head] = vs; ad1[wv * 4 + head] = vd; }
}

// ---------------- layer-1 segment max / sum over edges (+self loops) ----------------
__global__ void k_emax1(const int* __restrict__ ei, const float* __restrict__ as1,
                        const float* __restrict__ ad1, unsigned* __restrict__ emax,
                        int E, int N) {
  int i = blockIdx.x * blockDim.x + threadIdx.x;
  int T = E + N;
  if (i >= T) return;
  int s = (i < E) ? ei[i]     : (i - E);
  int d = (i < E) ? ei[E + i] : (i - E);
  float4 av = *(const float4*)(as1 + (long long)s * 4);
  float4 bv = *(const float4*)(ad1 + (long long)d * 4);
  atomicMax(&emax[d * 4 + 0], f2o(lrelu(av.x + bv.x)));
  atomicMax(&emax[d * 4 + 1], f2o(lrelu(av.y + bv.y)));
  atomicMax(&emax[d * 4 + 2], f2o(lrelu(av.z + bv.z)));
  atomicMax(&emax[d * 4 + 3], f2o(lrelu(av.w + bv.w)));
}

__global__ void k_denom1(const int* __restrict__ ei, const float* __restrict__ as1,
                         const float* __restrict__ ad1, const unsigned* __restrict__ emax,
                         float* __restrict__ denom, int E, int N) {
  int i = blockIdx.x * blockDim.x + threadIdx.x;
  int T = E + N;
  if (i >= T) return;
  int s = (i < E) ? ei[i]     : (i - E);
  int d = (i < E) ? ei[E + i] : (i - E);
  float4 av = *(const float4*)(as1 + (long long)s * 4);
  float4 bv = *(const float4*)(ad1 + (long long)d * 4);
  float e[4] = { lrelu(av.x + bv.x), lrelu(av.y + bv.y), lrelu(av.z + bv.z), lrelu(av.w + bv.w) };
#pragma unroll
  for (int h = 0; h < 4; ++h)
    atomicAdd(&denom[d * 4 + h], __expf(e[h] - o2f(emax[d * 4 + h])));
}

// ---------------- layer-1 message scatter: one wave per edge ----------------
__global__ void k_scatter1(const int* __restrict__ ei, const float* __restrict__ h1,
                           const float* __restrict__ as1, const float* __restrict__ ad1,
                           const unsigned* __restrict__ emax, const float* __restrict__ denom,
                           float* __restrict__ out1, int E, int N) {
  long long gid = (long long)blockIdx.x * blockDim.x + threadIdx.x;
  int w = (int)(gid >> 5);
  int lane = threadIdx.x & 31;
  int T = E + N;
  if (w >= T) return;
  int s = (w < E) ? ei[w]     : (w - E);
  int d = (w < E) ? ei[E + w] : (w - E);
  int head = lane >> 3;
  float e     = lrelu(as1[s * 4 + head] + ad1[d * 4 + head]);
  float ex    = __expf(e - o2f(emax[d * 4 + head]));
  float alpha = ex / (denom[d * 4 + head] + 1e-16f);
  int c = lane * 4;
  float4 hv = *(const float4*)(h1 + (long long)s * D1 + c);
  float* ob = out1 + (long long)d * D1 + c;
  atomicAdd(ob + 0, hv.x * alpha);
  atomicAdd(ob + 1, hv.y * alpha);
  atomicAdd(ob + 2, hv.z * alpha);
  atomicAdd(ob + 3, hv.w * alpha);
}

// ---------------- bias + leaky_relu (in place) ----------------
__global__ void k_bias_lrelu(float* __restrict__ p, const float* __restrict__ b,
                             long long total, int dim) {
  long long i = (long long)blockIdx.x * blockDim.x + threadIdx.x;
  if (i < total) p[i] = lrelu(p[i] + b[(int)(i % dim)]);
}

// ---------------- layer-2 attention coefficients (H=1, C=40) ----------------
__global__ void k_alpha2(const float* __restrict__ h2, const float* __restrict__ a_src,
                         const float* __restrict__ a_dst, float* __restrict__ as2,
                         float* __restrict__ ad2, int N) {
  int wv = (int)(((long long)blockIdx.x * blockDim.x + threadIdx.x) >> 5);
  int lane = threadIdx.x & 31;
  if (wv >= N) return;
  const float* row = h2 + (long long)wv * OUT_DIM;
  float x0 = row[lane];
  float vs = x0 * a_src[lane];
  float vd = x0 * a_dst[lane];
  if (lane < 8) {
    float x1 = row[32 + lane];
    vs += x1 * a_src[32 + lane];
    vd += x1 * a_dst[32 + lane];
  }
  for (int off = 1; off < 32; off <<= 1) { vs += __shfl_xor(vs, off); vd += __shfl_xor(vd, off); }
  if (lane == 0) { as2[wv] = vs; ad2[wv] = vd; }
}

__global__ void k_emax2(const int* __restrict__ ei, const float* __restrict__ as2,
                        const float* __restrict__ ad2, unsigned* __restrict__ emax,
                        int E, int N) {
  int i = blockIdx.x * blockDim.x + threadIdx.x;
  int T = E + N;
  if (i >= T) return;
  int s = (i < E) ? ei[i]     : (i - E);
  int d = (i < E) ? ei[E + i] : (i - E);
  atomicMax(&emax[d], f2o(lrelu(as2[s] + ad2[d])));
}

__global__ void k_denom2(const int* __restrict__ ei, const float* __restrict__ as2,
                         const float* __restrict__ ad2, const unsigned* __restrict__ emax,
                         float* __restrict__ denom, int E, int N) {
  int i = blockIdx.x * blockDim.x + threadIdx.x;
  int T = E + N;
  if (i >= T) return;
  int s = (i < E) ? ei[i]     : (i - E);
  int d = (i < E) ? ei[E + i] : (i - E);
  float e = lrelu(as2[s] + ad2[d]);
  atomicAdd(&denom[d], __expf(e - o2f(emax[d])));
}

__global__ void k_scatter2(const int* __restrict__ ei, const float* __restrict__ h2,
                           const float* __restrict__ as2, const float* __restrict__ ad2,
                           const unsigned* __restrict__ emax, const float* __restrict__ denom,
                           float* __restrict__ out2, int E, int N) {
  long long gid = (long long)blockIdx.x * blockDim.x + threadIdx.x;
  int w = (int)(gid >> 5);
  int lane = threadIdx.x & 31;
  int T = E + N;
  if (w >= T) return;
  int s = (w < E) ? ei[w]     : (w - E);
  int d = (w < E) ? ei[E + w] : (w - E);
  float e     = lrelu(as2[s] + ad2[d]);
  float alpha = __expf(e - o2f(emax[d])) / (denom[d] + 1e-16f);
  const float* hs = h2 + (long long)s * OUT_DIM;
  float* ob = out2 + (long long)d * OUT_DIM;
  atomicAdd(ob + lane, hs[lane] * alpha);
  if (lane < 8) atomicAdd(ob + 32 + lane, hs[32 + lane] * alpha);
}

// ---------------- z = out2 + b2 ; log_softmax ; write both outputs ----------------
__global__ void k_final(const float* __restrict__ out2, const float* __restrict__ b2,
                        float* __restrict__ dout, int N) {
  int wv = (int)(((long long)blockIdx.x * blockDim.x + threadIdx.x) >> 5);
  int lane = threadIdx.x & 31;
  if (wv >= N) return;
  const float* r = out2 + (long long)wv * OUT_DIM;
  float z0 = r[lane] + b2[lane];
  float z1 = (lane < 8) ? (r[32 + lane] + b2[32 + lane]) : -INFINITY;
  float m = fmaxf(z0, z1);
  for (int off = 1; off < 32; off <<= 1) m = fmaxf(m, __shfl_xor(m, off));
  float s = __expf(z0 - m) + ((lane < 8) ? __expf(z1 - m) : 0.f);
  for (int off = 1; off < 32; off <<= 1) s += __shfl_xor(s, off);
  float lse = m + __logf(s);
  long long base  = (long long)wv * OUT_DIM;
  long long base2 = (long long)N * OUT_DIM + base;
  dout[base + lane]  = z0;
  dout[base2 + lane] = z0 - lse;
  if (lane < 8) {
    dout[base + 32 + lane]  = z1;
    dout[base2 + 32 + lane] = z1 - lse;
  }
}

extern "C" void kernel_launch(void* const* d_in, const int* in_sizes, int n_in,
                              void* d_out, int out_size, void* d_ws, size_t ws_size,
                              hipStream_t stream) {
  (void)n_in; (void)out_size; (void)ws_size;
  const float* x      = (const float*)d_in[0];
  const int*   ei     = (const int*)d_in[1];
  const float* W1     = (const float*)d_in[2];
  const float* a_src1 = (const float*)d_in[3];
  const float* a_dst1 = (const float*)d_in[4];
  const float* b1     = (const float*)d_in[5];
  const float* W2     = (const float*)d_in[6];
  const float* a_src2 = (const float*)d_in[7];
  const float* a_dst2 = (const float*)d_in[8];
  const float* b2     = (const float*)d_in[9];
  float* dout = (float*)d_out;

  int N = in_sizes[0] / IN_DIM;
  int E = in_sizes[1] / 2;
  int T = E + N;

  float* ws = (float*)d_ws;
  long long off = 0;
  // ---- zeroed accumulator region (contiguous, at front) ----
  float*    out1   = ws + off; off += (long long)N * D1;
  unsigned* emax1  = (unsigned*)(ws + off); off += (long long)N * 4;
  float*    denom1 = ws + off; off += (long long)N * 4;
  float*    out2   = ws + off; off += (long long)N * OUT_DIM;
  unsigned* emax2  = (unsigned*)(ws + off); off += N;
  float*    denom2 = ws + off; off += N;
  long long zerof = off;
  // ---- fully-overwritten buffers (no init needed) ----
  float*    h1     = ws + off; off += (long long)N * D1;
  float*    as1    = ws + off; off += (long long)N * 4;
  float*    ad1    = ws + off; off += (long long)N * 4;
  float*    h2     = ws + off; off += (long long)N * OUT_DIM;
  float*    as2    = ws + off; off += N;
  float*    ad2    = ws + off; off += N;
  float*    W2p    = ws + off; off += (long long)D1 * OUT_PAD;

  k_zero<<<(unsigned)((zerof + 255) / 256), 256, 0, stream>>>(ws, zerof);
  k_padB<<<(D1 * OUT_PAD + 255) / 256, 256, 0, stream>>>(W2, W2p, D1, OUT_DIM, OUT_PAD);

  int rt = (N + 15) / 16;
  // layer 1 (W1 is already 16-col aligned: NCpad == NC == 128)
  k_gemm_wmma<IN_DIM, D1><<<rt, 256, 0, stream>>>(x, W1, h1, D1, N);
  k_alpha1<<<(unsigned)(((long long)N * 32 + 255) / 256), 256, 0, stream>>>(h1, a_src1, a_dst1, as1, ad1, N);
  k_emax1 <<<(T + 255) / 256, 256, 0, stream>>>(ei, as1, ad1, emax1, E, N);
  k_denom1<<<(T + 255) / 256, 256, 0, stream>>>(ei, as1, ad1, emax1, denom1, E, N);
  k_scatter1<<<(unsigned)(((long long)T * 32 + 255) / 256), 256, 0, stream>>>(ei, h1, as1, ad1, emax1, denom1, out1, E, N);
  {
    long long tot = (long long)N * D1;
    k_bias_lrelu<<<(unsigned)((tot + 255) / 256), 256, 0, stream>>>(out1, b1, tot, D1);
  }
  // layer 2 (3 waves per block cover 48 padded cols; stores guarded to 40)
  k_gemm_wmma<D1, OUT_PAD><<<rt, 96, 0, stream>>>(out1, W2p, h2, OUT_DIM, N);
  k_alpha2<<<(unsigned)(((long long)N * 32 + 255) / 256), 256, 0, stream>>>(h2, a_src2, a_dst2, as2, ad2, N);
  k_emax2 <<<(T + 255) / 256, 256, 0, stream>>>(ei, as2, ad2, emax2, E, N);
  k_denom2<<<(T + 255) / 256, 256, 0, stream>>>(ei, as2, ad2, emax2, denom2, E, N);
  k_scatter2<<<(unsigned)(((long long)T * 32 + 255) / 256), 256, 0, stream>>>(ei, h2, as2, ad2, emax2, denom2, out2, E, N);
  k_final<<<(unsigned)(((long long)N * 32 + 255) / 256), 256, 0, stream>>>(out2, b2, dout, N);
}